// LSTMModel_14053132992735
// MI455X (gfx1250) — compile-verified
//
#include <hip/hip_runtime.h>
#include <hip/hip_bf16.h>
#include <math.h>

// ---------------- CDNA5 WMMA / TDM types ----------------
typedef float v2f __attribute__((ext_vector_type(2)));
typedef float v8f __attribute__((ext_vector_type(8)));
typedef unsigned int u32x4 __attribute__((ext_vector_type(4)));
typedef int i32x4 __attribute__((ext_vector_type(4)));
typedef int i32x8 __attribute__((ext_vector_type(8)));

#define DIM    128
#define GATES  512          // 4*DIM
#define BSZ    32
#define SEQ    256
#define ROWS   (BSZ*SEQ)    // 8192
#define VOCAB  32000

#if __has_builtin(__builtin_amdgcn_tensor_load_to_lds)
#define HAVE_TDM 1
#else
#define HAVE_TDM 0
#endif

// fragment loader: tile of 16 rows, row-major with row stride 128 floats.
// Works for A (activations [M,128]) and for B^T (weights stored [N,128]).
// V_WMMA_F32_16X16X4_F32 layout: lanes 0-15 hold K=k0,k0+1; lanes 16-31 hold K=k0+2,k0+3.
__device__ __forceinline__ v2f load_frag(const float* base, int lane, int k0) {
    int r     = lane & 15;
    int khalf = (lane >> 4) << 1;          // 0 or 2
    return *(const v2f*)(base + r * DIM + k0 + khalf);
}

__device__ __forceinline__ float sigmoidf_(float x) { return 1.0f / (1.0f + __expf(-x)); }

// ---------------- embedding gather ----------------
__global__ __launch_bounds__(256) void embed_kernel(const int* __restrict__ x,
                                                    const float* __restrict__ emb,
                                                    float* __restrict__ h) {
    int i   = blockIdx.x * blockDim.x + threadIdx.x;   // over ROWS*32 float4
    int row = i >> 5;
    int c   = i & 31;
    int tok = x[row];
    ((float4*)h)[i] = ((const float4*)(emb + (size_t)tok * DIM))[c];
}

// ---------------- generic WMMA GEMM: C[M,N] = A[M,128] * W[N,128]^T + bias ----------------
// one wave computes a 16x64 tile: four 16x16 WMMA accumulators sharing one A fragment,
// so each K-step is 5 b64 loads feeding 4 v_wmma ops.
__global__ __launch_bounds__(256) void wmma_gemm_bias(const float* __restrict__ A,
                                                      const float* __restrict__ W,
                                                      const float* __restrict__ bias0,
                                                      const float* __restrict__ bias1, // may be null
                                                      float* __restrict__ C,
                                                      int M, int N) {
    const int wavesPerBlk = blockDim.x >> 5;
    const int w      = blockIdx.x * wavesPerBlk + (threadIdx.x >> 5);
    const int nquads = N >> 6;
    const int mt = w / nquads;
    const int nq = w - mt * nquads;
    if (mt * 16 >= M) return;                      // uniform per wave
    const int lane = threadIdx.x & 31;

    const float* Ab = A + (size_t)mt * 16 * DIM;
    const float* Wb = W + (size_t)(nq * 64) * DIM;

    v8f acc[4] = {};
    #pragma unroll 4
    for (int k0 = 0; k0 < DIM; k0 += 4) {
        v2f a  = load_frag(Ab, lane, k0);
        v2f b0 = load_frag(Wb +  0 * DIM, lane, k0);
        v2f b1 = load_frag(Wb + 16 * DIM, lane, k0);
        v2f b2 = load_frag(Wb + 32 * DIM, lane, k0);
        v2f b3 = load_frag(Wb + 48 * DIM, lane, k0);
        acc[0] = __builtin_amdgcn_wmma_f32_16x16x4_f32(false, a, false, b0, (short)0, acc[0], false, false);
        acc[1] = __builtin_amdgcn_wmma_f32_16x16x4_f32(false, a, false, b1, (short)0, acc[1], false, false);
        acc[2] = __builtin_amdgcn_wmma_f32_16x16x4_f32(false, a, false, b2, (short)0, acc[2], false, false);
        acc[3] = __builtin_amdgcn_wmma_f32_16x16x4_f32(false, a, false, b3, (short)0, acc[3], false, false);
    }

    const int rowbase = mt * 16 + ((lane >> 4) << 3);
    #pragma unroll
    for (int j = 0; j < 4; ++j) {
        const int n = nq * 64 + j * 16 + (lane & 15);
        const float bv = bias0[n] + (bias1 ? bias1[n] : 0.0f);
        #pragma unroll
        for (int r = 0; r < 8; ++r) {
            size_t row = (size_t)(rowbase + r);
            C[row * (size_t)N + n] = acc[j][r] + bv;
        }
    }
}

// ---------------- TDM staging: DMA W_hh [512,128] f32 from global into LDS ----------------
__device__ __forceinline__ void tdm_stage_whh(const float* __restrict__ Whh, float* sW) {
#if HAVE_TDM
    // D# group 0: count=1 (valid user descriptor), lds_addr, 57-bit global addr, type=2
    unsigned long long ga = (unsigned long long)(const void*)Whh;
    unsigned lds_off = (unsigned)(unsigned long long)(void*)sW;  // flat addr[31:0] == LDS offset
    u32x4 g0;
    g0[0] = 1u;
    g0[1] = lds_off;
    g0[2] = (unsigned)(ga & 0xFFFFFFFFu);
    g0[3] = (unsigned)((ga >> 32) & 0x01FFFFFFu) | (2u << 30);
    // D# group 1: data_size=2 (4B), tensor 128 x 512, tile 128 x 512, dim0 stride 128
    i32x8 g1;
    g1[0] = (int)(2u << 16);          // workgroup_mask=0, data_size=4B
    g1[1] = (int)(128u << 16);        // tensor_dim0[15:0] @ bit48
    g1[2] = (int)(512u << 16);        // tensor_dim1[15:0] @ bit80
    g1[3] = (int)(128u << 16);        // tile_dim0 @ bit112
    g1[4] = (int)512;                 // tile_dim1 @ bit128, tile_dim2=0
    g1[5] = (int)128;                 // tensor_dim0_stride[31:0]
    g1[6] = 0;
    g1[7] = 0;
    i32x4 gz4 = {0, 0, 0, 0};
    i32x8 gz8 = {0, 0, 0, 0, 0, 0, 0, 0};
    __builtin_amdgcn_tensor_load_to_lds(g0, g1, gz4, gz4, gz8, 0);
    __builtin_amdgcn_s_wait_tensorcnt((unsigned short)0);
#else
    const float4* Wg = (const float4*)Whh;
    float4*       Ws = (float4*)sW;
    for (int i = threadIdx.x; i < (GATES * DIM) / 4; i += 512) Ws[i] = Wg[i];
#endif
}

// ---------------- LSTM recurrent scan: one workgroup, W_hh + h resident in 320KB LDS ----------------
// 512 threads = 16 waves. wave -> (mt in 0..1 batch-tile, nt in 0..7 d-tile).
// Each wave owns ALL FOUR gate tiles for its (mt,nt): i/f/g/o mix in-registers,
// and cell state c lives in registers for the whole 256-step sequence.
__global__ __launch_bounds__(512) void lstm_scan_kernel(const float* __restrict__ xW,   // [B,S,4D]
                                                        const float* __restrict__ Whh,  // [4D,128]
                                                        float* __restrict__ h_out) {    // [B,S,128]
    extern __shared__ float smem[];
    float* sW = smem;                    // 512*128 floats (256 KB)
    float* sh = smem + GATES * DIM;      // 32*128 floats (16 KB)

    const int tid  = threadIdx.x;
    const int wave = tid >> 5;
    const int lane = tid & 31;

#if HAVE_TDM
    if (wave == 0) {                     // one wave issues the tensor DMA + waits TENSORcnt
        tdm_stage_whh(Whh, sW);
    }
#else
    tdm_stage_whh(Whh, sW);
#endif
    for (int i = tid; i < BSZ * DIM; i += 512) sh[i] = 0.0f;
    __syncthreads();

    const int mt   = wave & 1;          // batch tile 0..1
    const int nt   = wave >> 1;         // d tile 0..7
    const int rhi  = lane >> 4;
    const int col  = lane & 15;

    const float* aBase = sh + (mt * 16) * DIM;
    const float* wI = sW + (size_t)(0 * DIM + nt * 16) * DIM;
    const float* wF = sW + (size_t)(1 * DIM + nt * 16) * DIM;
    const float* wG = sW + (size_t)(2 * DIM + nt * 16) * DIM;
    const float* wO = sW + (size_t)(3 * DIM + nt * 16) * DIM;

    float creg[8];
    #pragma unroll
    for (int r = 0; r < 8; ++r) creg[r] = 0.0f;

    for (int t = 0; t < SEQ; ++t) {
        v8f ai = {}; v8f af = {}; v8f ag = {}; v8f ao = {};
        #pragma unroll 4
        for (int k0 = 0; k0 < DIM; k0 += 4) {
            v2f a  = load_frag(aBase, lane, k0);
            v2f bi = load_frag(wI,    lane, k0);
            v2f bf = load_frag(wF,    lane, k0);
            v2f bg = load_frag(wG,    lane, k0);
            v2f bo = load_frag(wO,    lane, k0);
            ai = __builtin_amdgcn_wmma_f32_16x16x4_f32(false, a, false, bi, (short)0, ai, false, false);
            af = __builtin_amdgcn_wmma_f32_16x16x4_f32(false, a, false, bf, (short)0, af, false, false);
            ag = __builtin_amdgcn_wmma_f32_16x16x4_f32(false, a, false, bg, (short)0, ag, false, false);
            ao = __builtin_amdgcn_wmma_f32_16x16x4_f32(false, a, false, bo, (short)0, ao, false, false);
        }
        __syncthreads();   // everyone done reading old h from LDS

        #pragma unroll
        for (int r = 0; r < 8; ++r) {
            int b = mt * 16 + (rhi << 3) + r;       // C/D layout: M = r + 8*(lane>=16)
            int d = nt * 16 + col;
            const float* xw = xW + (((size_t)b * SEQ + t) * GATES) + d;
            float gi = sigmoidf_(ai[r] + xw[0 * DIM]);
            float gf = sigmoidf_(af[r] + xw[1 * DIM]);
            float gg = tanhf   (ag[r] + xw[2 * DIM]);
            float go = sigmoidf_(ao[r] + xw[3 * DIM]);
            float c  = gf * creg[r] + gi * gg;
            creg[r]  = c;
            float h  = go * tanhf(c);
            sh[b * DIM + d] = h;
            h_out[((size_t)b * SEQ + t) * DIM + d] = h;
        }
        __syncthreads();   // new h visible before next step's matmul
    }
}

// ---------------- LayerNorm over last dim (D=128), one wave per row ----------------
__global__ __launch_bounds__(256) void ln_kernel(const float* __restrict__ h,
                                                 const float* __restrict__ gamma,
                                                 const float* __restrict__ beta,
                                                 float* __restrict__ out) {
    int wave = threadIdx.x >> 5;
    int lane = threadIdx.x & 31;
    int row  = blockIdx.x * 8 + wave;
    float4 v = ((const float4*)(h + (size_t)row * DIM))[lane];
    float s  = v.x + v.y + v.z + v.w;
    float sq = v.x * v.x + v.y * v.y + v.z * v.z + v.w * v.w;
    #pragma unroll
    for (int off = 16; off > 0; off >>= 1) {
        s  += __shfl_xor(s,  off);
        sq += __shfl_xor(sq, off);
    }
    float mu   = s * (1.0f / DIM);
    float var  = sq * (1.0f / DIM) - mu * mu;
    float rstd = rsqrtf(var + 1e-5f);
    float4 g = ((const float4*)gamma)[lane];
    float4 b = ((const float4*)beta)[lane];
    float4 o;
    o.x = (v.x - mu) * rstd * g.x + b.x;
    o.y = (v.y - mu) * rstd * g.y + b.y;
    o.z = (v.z - mu) * rstd * g.z + b.z;
    o.w = (v.w - mu) * rstd * g.w + b.w;
    ((float4*)(out + (size_t)row * DIM))[lane] = o;
}

// ---------------- host launch ----------------
extern "C" void kernel_launch(void* const* d_in, const int* in_sizes, int n_in,
                              void* d_out, int out_size, void* d_ws, size_t ws_size,
                              hipStream_t stream) {
    const int*   x     = (const int*)  d_in[0];
    const float* emb   = (const float*)d_in[1];
    const float* W_ih  = (const float*)d_in[2];   // [4,512,128]
    const float* W_hh  = (const float*)d_in[3];   // [4,512,128]
    const float* b_ih  = (const float*)d_in[4];   // [4,512]
    const float* b_hh  = (const float*)d_in[5];   // [4,512]
    const float* gamma = (const float*)d_in[6];
    const float* beta  = (const float*)d_in[7];
    const float* headW = (const float*)d_in[8];   // [32000,128]
    const float* headb = (const float*)d_in[9];
    float*       out   = (float*)d_out;

    float* hA = (float*)d_ws;                // ROWS*DIM
    float* hB = hA + (size_t)ROWS * DIM;     // ROWS*DIM
    float* xW = hB + (size_t)ROWS * DIM;     // ROWS*GATES

    // 1) embedding gather
    embed_kernel<<<(ROWS * 32) / 256, 256, 0, stream>>>(x, emb, hA);

    const size_t scanLds = (size_t)(GATES * DIM + BSZ * DIM) * sizeof(float); // 272 KB < 320 KB/WGP
    (void)hipFuncSetAttribute((const void*)lstm_scan_kernel,
                              hipFuncAttributeMaxDynamicSharedMemorySize, (int)scanLds);

    float* hcur = hA;
    float* hnxt = hB;
    for (int l = 0; l < 4; ++l) {
        // xW = h @ W_ih^T + b_ih + b_hh : M=8192(512 mtiles), N=512(8 nquads)
        wmma_gemm_bias<<<(512 * 8) / 8, 256, 0, stream>>>(
            hcur, W_ih + (size_t)l * GATES * DIM,
            b_ih + (size_t)l * GATES, b_hh + (size_t)l * GATES,
            xW, ROWS, GATES);
        // recurrence: single workgroup, weights staged into LDS by the Tensor Data Mover
        lstm_scan_kernel<<<1, 512, scanLds, stream>>>(
            xW, W_hh + (size_t)l * GATES * DIM, hnxt);
        float* tmp = hcur; hcur = hnxt; hnxt = tmp;
    }

    // 2) LayerNorm -> hnxt
    ln_kernel<<<ROWS / 8, 256, 0, stream>>>(hcur, gamma, beta, hnxt);

    // 3) head: out = hn @ headW^T + head_b : M=8192(512 mtiles), N=32000(500 nquads)
    wmma_gemm_bias<<<(512 * 500) / 8, 256, 0, stream>>>(
        hnxt, headW, headb, nullptr, out, ROWS, VOCAB);
}